// RMSDyadReLUBlock_52080773431507
// MI455X (gfx1250) — compile-verified
//
#include <hip/hip_runtime.h>
#include <hip/hip_bf16.h>

// RMS-norm + dyadic block GEMM + ReLU for MI455X (gfx1250).
// fp32 WMMA (V_WMMA_F32_16X16X4_F32) + async global->LDS staging (ASYNCcnt).
//
// y[d*256+o, c] = relu( fc[c] * sum_i( Wup[d,o,i]*g[d*256+i]*x[d*256+i, c]
//                                    + Wlo[d,o,i]*g[(15-d)*256+i]*x[(15-d)*256+i, c] )
//                       + bias[d*256+o] )
// fc[c] = rsqrt( mean_r x[r,c]^2 + eps )
//
// rms_params (g) folded into weights during the reg->LDS stage; fc in epilogue.
// x-tile stored TRANSPOSED in LDS so B-fragments are single ds_load_b64 reads.

#define DY    16
#define DOUT  256
#define DIN   256
#define BS    4096
#define NFEAT (DY * DIN)   // 4096
#define EPSV  1e-6f

#define BM 128   // M tile per CTA
#define BN 128   // N tile per CTA
#define KC 32    // K chunk staged in LDS
#define PW 36    // sW pitch (floats): A-frag b64 reads conflict-free
#define PT 34    // sXt pitch (floats): 34*l mod 64 distinct for 32 lanes -> scatter conflict-free
#define XBUF (BN * PT)

typedef float v2f __attribute__((ext_vector_type(2)));
typedef float v8f __attribute__((ext_vector_type(8)));

// ---------------- Phase 1a: zero the per-column sum buffer ----------------
__global__ void zero_sums_kernel(float* __restrict__ sums) {
    int i = blockIdx.x * 256 + threadIdx.x;
    if (i < BS) sums[i] = 0.0f;
}

// ---------------- Phase 1b: column sum of squares (partial + atomic) ------
__global__ void colsumsq_kernel(const float* __restrict__ x,
                                float* __restrict__ sums) {
    const int col = blockIdx.x * 256 + threadIdx.x;
    const int r0  = blockIdx.y * 256;
    float s = 0.0f;
    #pragma unroll 8
    for (int r = 0; r < 256; ++r) {
        float v = x[(size_t)(r0 + r) * BS + col];     // coalesced
        s += v * v;
    }
    atomicAdd(&sums[col], s);
}

// ---------------- Phase 1c: fc = rsqrt(mean + eps) ------------------------
__global__ void make_fc_kernel(const float* __restrict__ sums,
                               float* __restrict__ fc) {
    int i = blockIdx.x * 256 + threadIdx.x;
    if (i < BS) fc[i] = rsqrtf(sums[i] * (1.0f / (float)NFEAT) + EPSV);
}

// -------- async transposing copy of one KC x BN x-tile into LDS -----------
// 16 x GLOBAL_LOAD_ASYNC_TO_LDS_B32 per wave per chunk (ASYNCcnt tracked).
// Global side: 32 lanes read 128 contiguous bytes (consecutive n, fixed k).
// LDS side: per-lane scatter to sXt[n*PT + k] (transposed layout).
__device__ __forceinline__ void async_x_tile(const float* __restrict__ x,
                                             int rowbase, int c0,
                                             float* ldsbase, int t) {
    #pragma unroll
    for (int p = 0; p < 16; ++p) {
        const int e = p * 256 + t;            // 0..4095 elements
        const int k = e >> 7;                 // 0..31
        const int n = e & 127;                // 0..127
        const float* gsrc = x + (size_t)(rowbase + k) * BS + c0 + n;
        const unsigned ldst = (unsigned)(size_t)(ldsbase + n * PT + k);
        asm volatile("global_load_async_to_lds_b32 %0, %1, off"
                     :: "v"(ldst), "v"((unsigned long long)(size_t)gsrc)
                     : "memory");
    }
}

// -------- prefetch one BM x KC weight tile (+ g chunk) into registers -----
__device__ __forceinline__ void load_w_regs(const float* __restrict__ wsrc,
                                            const float* __restrict__ g,
                                            int rowbase, int o0, int kloc, int t,
                                            float4* wv, float4* gv) {
    #pragma unroll
    for (int p = 0; p < 4; ++p) {
        const int o  = (t >> 3) + 32 * p;     // 0..127
        const int c4 = (t & 7) * 4;           // 0..28
        wv[p] = *(const float4*)(wsrc + (size_t)(o0 + o) * DIN + kloc + c4);
        gv[p] = *(const float4*)(g + rowbase + c4);
    }
}

// ---------------- Phase 2: dyadic GEMM + bias + ReLU ----------------------
__global__ __launch_bounds__(256)
void dyad_gemm_kernel(const float* __restrict__ x,
                      const float* __restrict__ g,
                      const float* __restrict__ wu,
                      const float* __restrict__ wl,
                      const float* __restrict__ bias,
                      const float* __restrict__ fc,
                      float* __restrict__ out) {
    __shared__ float sW[BM * PW];        // g-scaled weights [o][k] (single buffer)
    __shared__ float sXt[2 * XBUF];      // x tile TRANSPOSED [n][k], double-buffered

    const int d    = blockIdx.z;
    const int o0   = blockIdx.y * BM;
    const int c0   = blockIdx.x * BN;
    const int t    = threadIdx.x;
    const int wave = t >> 5;             // 8 waves, wave32
    const int lane = t & 31;
    const int lo16 = lane & 15;
    const int hi   = lane >> 4;
    const int wm   = wave >> 1;          // 4 M strips of 32 rows
    const int wn   = wave & 1;           // 2 N strips of 64 cols

    // Wave tile 32x64 -> 2 (am) x 4 (bn) accumulators of 16x16.
    v8f acc[8];
    #pragma unroll
    for (int i = 0; i < 8; ++i)
        #pragma unroll
        for (int j = 0; j < 8; ++j) acc[i][j] = 0.0f;

    // chunk kb: 0..7 -> (w_upper, block d); 8..15 -> (w_lower, block 15-d)
    auto chunk_rowbase = [&](int kb) {
        return ((kb < 8) ? d : (DY - 1 - d)) * DIN + (kb & 7) * KC;
    };
    auto chunk_wsrc = [&](int kb) {
        return ((kb < 8) ? wu : wl) + (size_t)d * DOUT * DIN;
    };

    // ---- prologue: async x-tile 0, prefetch W-tile 0 into registers ----
    float4 wv[4], gv[4];
    async_x_tile(x, chunk_rowbase(0), c0, &sXt[0], t);
    load_w_regs(chunk_wsrc(0), g, chunk_rowbase(0), o0, 0, t, wv, gv);

    for (int kb = 0; kb < 16; ++kb) {
        const int cur = kb & 1;
        float4 wvn[4], gvn[4];

        if (kb + 1 < 16) {
            // overlap next chunk's traffic with this chunk's compute
            async_x_tile(x, chunk_rowbase(kb + 1), c0, &sXt[(cur ^ 1) * XBUF], t);
            load_w_regs(chunk_wsrc(kb + 1), g, chunk_rowbase(kb + 1),
                        o0, ((kb + 1) & 7) * KC, t, wvn, gvn);
        }

        // stage current W tile (scaled by rms_params) into LDS
        #pragma unroll
        for (int p = 0; p < 4; ++p) {
            const int o  = (t >> 3) + 32 * p;
            const int c4 = (t & 7) * 4;
            float* dst = &sW[o * PW + c4];
            dst[0] = wv[p].x * gv[p].x; dst[1] = wv[p].y * gv[p].y;
            dst[2] = wv[p].z * gv[p].z; dst[3] = wv[p].w * gv[p].w;
        }

        // retire the async group for the CURRENT buffer (async loads complete
        // in order: <=16 outstanding leaves only the group just issued for kb+1)
        if (kb + 1 < 16) asm volatile("s_wait_asynccnt 16" ::: "memory");
        else             asm volatile("s_wait_asynccnt 0" ::: "memory");
        __syncthreads();

        const float* xb = &sXt[cur * XBUF];
        #pragma unroll
        for (int kk = 0; kk < KC; kk += 4) {
            const int ka = kk + 2 * hi;               // fp32 A/B ISA layout: hi half -> K+2
            v2f a0 = *(const v2f*)&sW[(wm * 32 +      lo16) * PW + ka];
            v2f a1 = *(const v2f*)&sW[(wm * 32 + 16 + lo16) * PW + ka];
            #pragma unroll
            for (int bn = 0; bn < 4; ++bn) {
                const int n = wn * 64 + bn * 16 + lo16;
                v2f b = *(const v2f*)&xb[n * PT + ka];   // contiguous (k, k+1)
                acc[bn] = __builtin_amdgcn_wmma_f32_16x16x4_f32(
                    false, a0, false, b, (short)0, acc[bn], false, false);
                acc[4 + bn] = __builtin_amdgcn_wmma_f32_16x16x4_f32(
                    false, a1, false, b, (short)0, acc[4 + bn], false, false);
            }
        }
        __syncthreads();   // guard sW overwrite next iteration

        #pragma unroll
        for (int p = 0; p < 4; ++p) { wv[p] = wvn[p]; gv[p] = gvn[p]; }
    }

    // ---- epilogue: y = relu(fc[c] * acc + bias[row]) ----
    #pragma unroll
    for (int am = 0; am < 2; ++am) {
        float bv[8];
        #pragma unroll
        for (int v = 0; v < 8; ++v)
            bv[v] = bias[d * DOUT + o0 + wm * 32 + am * 16 + v + 8 * hi];
        #pragma unroll
        for (int bn = 0; bn < 4; ++bn) {
            const int c = c0 + wn * 64 + bn * 16 + lo16;
            const float f = fc[c];
            #pragma unroll
            for (int v = 0; v < 8; ++v) {
                const int row = d * DOUT + o0 + wm * 32 + am * 16 + v + 8 * hi;
                float val = acc[am * 4 + bn][v] * f + bv[v];
                val = fmaxf(val, 0.0f);
                out[(size_t)row * BS + c] = val;
            }
        }
    }
}

extern "C" void kernel_launch(void* const* d_in, const int* in_sizes, int n_in,
                              void* d_out, int out_size, void* d_ws, size_t ws_size,
                              hipStream_t stream) {
    (void)in_sizes; (void)n_in; (void)out_size; (void)ws_size;
    const float* x    = (const float*)d_in[0];
    const float* g    = (const float*)d_in[1];   // rms_params
    const float* wu   = (const float*)d_in[2];
    const float* wl   = (const float*)d_in[3];
    const float* bias = (const float*)d_in[4];
    float* out  = (float*)d_out;
    float* sums = (float*)d_ws;        // 4096 floats
    float* fc   = sums + BS;           // 4096 floats

    zero_sums_kernel<<<BS / 256, 256, 0, stream>>>(sums);
    colsumsq_kernel<<<dim3(BS / 256, NFEAT / 256), 256, 0, stream>>>(x, sums);
    make_fc_kernel<<<BS / 256, 256, 0, stream>>>(sums, fc);
    dyad_gemm_kernel<<<dim3(BS / BN, DOUT / BM, DY), 256, 0, stream>>>(
        x, g, wu, wl, bias, fc, out);
}